// DenseOTPE_30940944400973
// MI455X (gfx1250) — compile-verified
//
#include <hip/hip_runtime.h>
#include <math.h>

// ---------------------------------------------------------------------------
// DenseOTPE fused kernels for MI455X (gfx1250, wave32)
//  - GEMM h = x@W via V_WMMA_F32_16X16X4_F32 (f32 precision preserved)
//  - 512MB streaming trace update is the real cost: NT loads/stores, float4
// ---------------------------------------------------------------------------

typedef __attribute__((ext_vector_type(2))) float v2f;
typedef __attribute__((ext_vector_type(4))) float v4f;
typedef __attribute__((ext_vector_type(8))) float v8f;

#define BSZ  32
#define DIN  1024
#define DOUT 1024

// flat offsets into d_out (return order: s, E_W2, E_b2, Rh_W2, Rh_b2, g_bar2, r2)
static constexpr size_t OFF_S   = 0;
static constexpr size_t OFF_EW  = (size_t)BSZ * DOUT;                       // 32768
static constexpr size_t OFF_EB  = OFF_EW  + (size_t)BSZ * DIN * DOUT;       // 33587200
static constexpr size_t OFF_RHW = OFF_EB  + (size_t)BSZ * DOUT;             // 33619968
static constexpr size_t OFF_RHB = OFF_RHW + (size_t)BSZ * DIN * DOUT;       // 67174400
static constexpr size_t OFF_GB  = OFF_RHB + (size_t)BSZ * DOUT;             // 67207168
static constexpr size_t OFF_R2  = OFF_GB  + (size_t)BSZ * DOUT;             // 67239936

#define BETA 0.9f
#define THR  1.0f
#define LEAK 0.9f

// ---------------------------------------------------------------------------
// Kernel 1: one wave32 per 16x16 output tile of h = x@W.
//   grid = (B/16) * (DOUT/16) = 2 * 64 = 128 blocks of 32 threads.
// A fragment (16x4 f32, ISA 7.12.2): lane L -> m = L%16, half = L/16,
//   vgpr v holds A[m][k0 + 2*half + v]           (v in {0,1})
// B fragment (4x16 f32): lane L -> n = L%16,
//   vgpr v holds B[k0 + 2*half + v][n]
// C/D (16x16 f32): vgpr v, lane L -> row = v + 8*half, col = L%16
// Epilogue computes the full (B,DOUT) spike/trace chain at each (row,col).
// ---------------------------------------------------------------------------
__global__ __launch_bounds__(32)
void dense_otpe_wmma_spike(const float* __restrict__ x,
                           const float* __restrict__ W,
                           const float* __restrict__ bias,
                           const float* __restrict__ u,
                           const float* __restrict__ E_b,
                           const float* __restrict__ Rh_b,
                           const float* __restrict__ g_bar,
                           const float* __restrict__ r,
                           float* __restrict__ out,
                           float* __restrict__ ws_dsdu,
                           float* __restrict__ ws_sg)
{
    const int tile     = blockIdx.x;        // 0..127
    const int row_tile = tile >> 6;         // 0..1
    const int col_tile = tile & 63;         // 0..63
    const int lane     = threadIdx.x;       // 0..31
    const int half     = lane >> 4;         // 0 or 1
    const int mn       = lane & 15;         // M for A, N for B/C/D

    const int row0 = row_tile * 16;
    const int col0 = col_tile * 16;

    const float* xrow = x + (size_t)(row0 + mn) * DIN;   // A row m
    const float* wcol = W + (size_t)col0 + mn;           // B column n

    v8f acc = {};
    for (int k0 = 0; k0 < DIN; k0 += 4) {
        const int ka = k0 + 2 * half;
        v2f a, bm;
        a.x  = xrow[ka];
        a.y  = xrow[ka + 1];
        bm.x = wcol[(size_t)ka * DOUT];
        bm.y = wcol[(size_t)(ka + 1) * DOUT];
        // D = A*B + C, f32 in / f32 accumulate (matches f32 reference exactly)
        acc = __builtin_amdgcn_wmma_f32_16x16x4_f32(
            /*neg_a=*/false, a, /*neg_b=*/false, bm,
            /*c_mod=*/(short)0, acc, /*reuse_a=*/false, /*reuse_b=*/false);
    }

    // Epilogue: per-lane C/D elements -> spike + small trace chains
    #pragma unroll
    for (int v = 0; v < 8; ++v) {
        const int row = row0 + v + 8 * half;     // batch index
        const int col = col0 + mn;               // output-channel index
        const size_t idx = (size_t)row * DOUT + col;

        const float h   = acc[v] + bias[col];
        const float un  = BETA * u[idx] + h;
        const float s   = 1.0f / (1.0f + __expf(-(un - THR)));
        const float sg  = s * (1.0f - s);
        const float a   = BETA * sg;             // ds_du

        const float Eb1  = BETA * E_b[idx] + 1.0f;
        const float dthb = a * Eb1 + sg;
        const float Eb2  = BETA * Eb1 + 1.0f;
        const float Rhb2 = a * Rh_b[idx] + dthb;

        float ratio = LEAK * r[row];
        const float r2 = ratio + 1.0f;
        ratio = ratio / r2;
        const float g2 = ratio * g_bar[idx] + (1.0f - ratio) * a;

        out[OFF_S   + idx] = s;
        out[OFF_EB  + idx] = Eb2;
        out[OFF_RHB + idx] = Rhb2;
        out[OFF_GB  + idx] = g2;
        ws_dsdu[idx] = a;
        ws_sg[idx]   = sg;
    }
}

// ---------------------------------------------------------------------------
// Kernel 2: r2 = LEAK*r + 1   (B values)
// ---------------------------------------------------------------------------
__global__ __launch_bounds__(32)
void dense_otpe_r2(const float* __restrict__ r, float* __restrict__ out)
{
    const int t = threadIdx.x;
    if (t < BSZ) out[OFF_R2 + t] = LEAK * r[t] + 1.0f;
}

// ---------------------------------------------------------------------------
// Kernel 3: big streaming trace update over (B, DIN, DOUT).
//   E1  = BETA*E_W + x
//   dth = ds_du*E1 + x*sg
//   E2  = BETA*E1 + x
//   Rh2 = ds_du*Rh_W + dth
// 512 MB of traffic -> nontemporal B128 loads/stores for the 4 big streams;
// the per-batch ds_du/sg rows (reused by 1024 blocks) use regular caching.
// One block per (b,i) row: 256 threads x float4 covers DOUT=1024.
// ---------------------------------------------------------------------------
__global__ __launch_bounds__(256)
void dense_otpe_trace(const float* __restrict__ x,
                      const float* __restrict__ E_W,
                      const float* __restrict__ Rh_W,
                      const float* __restrict__ ds_du,
                      const float* __restrict__ sg,
                      float* __restrict__ E_out,
                      float* __restrict__ Rh_out)
{
    const size_t rowblk = blockIdx.x;            // b*DIN + i, 0..32767
    const int    b      = (int)(rowblk >> 10);
    const float  xv     = x[rowblk];             // x[b][i], wave-uniform
    const size_t base   = rowblk * DOUT;
    const int    j      = threadIdx.x * 4;

    const v4f e  = __builtin_nontemporal_load((const v4f*)(E_W  + base + j));
    const v4f rh = __builtin_nontemporal_load((const v4f*)(Rh_W + base + j));
    const v4f a4 = *(const v4f*)(ds_du + (size_t)b * DOUT + j);
    const v4f s4 = *(const v4f*)(sg    + (size_t)b * DOUT + j);

    v4f e2, rh2;
    #pragma unroll
    for (int c = 0; c < 4; ++c) {
        const float E1  = BETA * e[c] + xv;
        const float dth = a4[c] * E1 + xv * s4[c];
        e2[c]  = BETA * E1 + xv;
        rh2[c] = a4[c] * rh[c] + dth;
    }
    __builtin_nontemporal_store(e2,  (v4f*)(E_out  + base + j));
    __builtin_nontemporal_store(rh2, (v4f*)(Rh_out + base + j));
}

// ---------------------------------------------------------------------------
extern "C" void kernel_launch(void* const* d_in, const int* in_sizes, int n_in,
                              void* d_out, int out_size, void* d_ws, size_t ws_size,
                              hipStream_t stream)
{
    const float* x     = (const float*)d_in[0];
    const float* W     = (const float*)d_in[1];
    const float* bias  = (const float*)d_in[2];
    const float* u     = (const float*)d_in[3];
    const float* E_W   = (const float*)d_in[4];
    const float* E_b   = (const float*)d_in[5];
    const float* Rh_W  = (const float*)d_in[6];
    const float* Rh_b  = (const float*)d_in[7];
    const float* g_bar = (const float*)d_in[8];
    const float* r     = (const float*)d_in[9];

    float* out     = (float*)d_out;
    float* ws_dsdu = (float*)d_ws;                       // (B, DOUT)
    float* ws_sg   = ws_dsdu + (size_t)BSZ * DOUT;       // (B, DOUT)

    // 1) WMMA GEMM + spike/small-trace epilogue (writes s, E_b2, Rh_b2, g_bar2, ws)
    dense_otpe_wmma_spike<<<dim3(128), dim3(32), 0, stream>>>(
        x, W, bias, u, E_b, Rh_b, g_bar, r, out, ws_dsdu, ws_sg);

    // 2) r2
    dense_otpe_r2<<<dim3(1), dim3(32), 0, stream>>>(r, out);

    // 3) Big bandwidth-bound trace update (E_W2, Rh_W2)
    dense_otpe_trace<<<dim3(BSZ * DIN), dim3(256), 0, stream>>>(
        x, E_W, Rh_W, ws_dsdu, ws_sg,
        out + OFF_EW, out + OFF_RHW);
}